// GATSegmentationModel_962072674894
// MI455X (gfx1250) — compile-verified
//
#include <hip/hip_runtime.h>
#include <hip/hip_bf16.h>
#include <math.h>

#define N_NODES 131072
#define N_EDGES 393216
#define E2      (N_EDGES + N_NODES)   // edges + self loops = 524288

typedef float v2f __attribute__((ext_vector_type(2)));
typedef float v8f __attribute__((ext_vector_type(8)));

// ---- monotone float <-> uint mapping for atomic max over floats ----
__device__ __forceinline__ unsigned fmap(float f) {
    unsigned u = __float_as_uint(f);
    return u ^ ((unsigned)((int)u >> 31) | 0x80000000u);
}
__device__ __forceinline__ float funmap(unsigned v) {
    unsigned u = (v & 0x80000000u) ? (v ^ 0x80000000u) : ~v;
    return __uint_as_float(u);
}
__device__ __forceinline__ float lrelu(float v) { return fmaxf(v, 0.2f * v); }

// K0: c_s[h] = sum_d W1[h*64+d]*a_src1[h*64+d]; c_d[h] likewise.  8 floats -> consts
__global__ void k_precompute(const float* __restrict__ W1,
                             const float* __restrict__ as1,
                             const float* __restrict__ ad1,
                             float* __restrict__ consts) {
    __shared__ float ss[256], sd[256];
    int t = threadIdx.x;
    float w = W1[t];
    ss[t] = w * as1[t];
    sd[t] = w * ad1[t];
    __syncthreads();
    if (t < 4) {
        float a = 0.f, b = 0.f;
        for (int d = 0; d < 64; ++d) { a += ss[t * 64 + d]; b += sd[t * 64 + d]; }
        consts[t]     = a;   // c_s[h]
        consts[4 + t] = b;   // c_d[h]
    }
}

// K1: layer-1 segment max of attention logits (4 heads)
__global__ void k_edge_max1(const int* __restrict__ ei, const float* __restrict__ x,
                            const float* __restrict__ consts, unsigned* __restrict__ m1) {
    int e = blockIdx.x * blockDim.x + threadIdx.x;
    if (e >= E2) return;
    int s, d;
    if (e < N_EDGES) { s = ei[e]; d = ei[N_EDGES + e]; } else { s = d = e - N_EDGES; }
    float xs = x[s], xd = x[d];
#pragma unroll
    for (int h = 0; h < 4; ++h) {
        float v = lrelu(xs * consts[h] + xd * consts[4 + h]);
        atomicMax(&m1[d * 4 + h], fmap(v));
    }
}

// K2: layer-1 softmax denominator + weighted numerator  S_num = sum exp(e-m)*x[src]
__global__ void k_edge_sum1(const int* __restrict__ ei, const float* __restrict__ x,
                            const float* __restrict__ consts,
                            const unsigned* __restrict__ m1,
                            float* __restrict__ den1, float* __restrict__ num1) {
    int e = blockIdx.x * blockDim.x + threadIdx.x;
    if (e >= E2) return;
    int s, d;
    if (e < N_EDGES) { s = ei[e]; d = ei[N_EDGES + e]; } else { s = d = e - N_EDGES; }
    float xs = x[s], xd = x[d];
#pragma unroll
    for (int h = 0; h < 4; ++h) {
        float v = lrelu(xs * consts[h] + xd * consts[4 + h]);
        float w = __expf(v - funmap(m1[d * 4 + h]));
        atomicAdd(&den1[d * 4 + h], w);
        atomicAdd(&num1[d * 4 + h], w * xs);
    }
}

// K4: per 16-node tile, h2[n] = sum_c relu(S[n,h(c)]*W1[c] + b1[c]) * W2[c]
// via V_WMMA_F32_16X16X4_F32: A = S[16x4], B = head-masked W1 column block [4x16].
__global__ void k_h2_wmma(const float* __restrict__ num1, const float* __restrict__ den1,
                          const float* __restrict__ W1, const float* __restrict__ b1,
                          const float* __restrict__ W2, float* __restrict__ h2) {
    int lane = threadIdx.x & 31;
    int wave = threadIdx.x >> 5;
    int base = blockIdx.x * 128 + wave * 16;   // 8 waves * 16 nodes per block
    int m    = lane & 15;                      // row (node) for lanes, col for B
    int khi  = lane >> 4;                      // 0: K={0,1}; 1: K={2,3}
    int node = base + m;
    int ha = 2 * khi, hb2 = 2 * khi + 1;

    v2f A;
    A.x = num1[node * 4 + ha]  / (den1[node * 4 + ha]  + 1e-16f);
    A.y = num1[node * 4 + hb2] / (den1[node * 4 + hb2] + 1e-16f);

    float acc[8];
#pragma unroll
    for (int r = 0; r < 8; ++r) acc[r] = 0.f;

#pragma unroll
    for (int j = 0; j < 16; ++j) {           // 16 column blocks of 16 (=256 cols)
        int col = j * 16 + m;                // this lane's column within the block
        float w  = W1[col];
        float bb = b1[col];
        float w2 = W2[col];
        int hblk = j >> 2;                   // head owning this column block
        v2f B;
        B.x = (ha  == hblk) ? w : 0.f;
        B.y = (hb2 == hblk) ? w : 0.f;
        v8f C = {};
        C = __builtin_amdgcn_wmma_f32_16x16x4_f32(
                /*neg_a=*/false, A, /*neg_b=*/false, B,
                /*c_mod=*/(short)0, C, /*reuse_a=*/false, /*reuse_b=*/false);
#pragma unroll
        for (int r = 0; r < 8; ++r) {
            float v = fmaxf(C[r] + bb, 0.f); // bias + ReLU
            acc[r] += v * w2;                // scale by W2, partial col-sum
        }
    }
    // reduce across the 16 lanes of each half (columns 0..15)
#pragma unroll
    for (int r = 0; r < 8; ++r) {
        float a = acc[r];
        a += __shfl_xor(a, 1, 32);
        a += __shfl_xor(a, 2, 32);
        a += __shfl_xor(a, 4, 32);
        a += __shfl_xor(a, 8, 32);
        acc[r] = a;
    }
    if (m == 0) {                            // lane 0 -> rows 0..7, lane 16 -> rows 8..15
#pragma unroll
        for (int r = 0; r < 8; ++r) h2[base + khi * 8 + r] = acc[r];
    }
}

// K5: layer-2 segment max (H=1)
__global__ void k_edge_max2(const int* __restrict__ ei, const float* __restrict__ h2,
                            const float* __restrict__ as2, const float* __restrict__ ad2,
                            unsigned* __restrict__ m2) {
    int e = blockIdx.x * blockDim.x + threadIdx.x;
    if (e >= E2) return;
    int s, d;
    if (e < N_EDGES) { s = ei[e]; d = ei[N_EDGES + e]; } else { s = d = e - N_EDGES; }
    float v = lrelu(h2[s] * as2[0] + h2[d] * ad2[0]);
    atomicMax(&m2[d], fmap(v));
}

// K6: layer-2 softmax sums
__global__ void k_edge_sum2(const int* __restrict__ ei, const float* __restrict__ h2,
                            const float* __restrict__ as2, const float* __restrict__ ad2,
                            const unsigned* __restrict__ m2,
                            float* __restrict__ den2, float* __restrict__ num2) {
    int e = blockIdx.x * blockDim.x + threadIdx.x;
    if (e >= E2) return;
    int s, d;
    if (e < N_EDGES) { s = ei[e]; d = ei[N_EDGES + e]; } else { s = d = e - N_EDGES; }
    float hs = h2[s];
    float v = lrelu(hs * as2[0] + h2[d] * ad2[0]);
    float w = __expf(v - funmap(m2[d]));
    atomicAdd(&den2[d], w);
    atomicAdd(&num2[d], w * hs);
}

// K7: out[n] = num2/(den2+eps) + b2
__global__ void k_final(const float* __restrict__ num2, const float* __restrict__ den2,
                        const float* __restrict__ b2, float* __restrict__ out) {
    int n = blockIdx.x * blockDim.x + threadIdx.x;
    if (n >= N_NODES) return;
    out[n] = num2[n] / (den2[n] + 1e-16f) + b2[0];
}

extern "C" void kernel_launch(void* const* d_in, const int* in_sizes, int n_in,
                              void* d_out, int out_size, void* d_ws, size_t ws_size,
                              hipStream_t stream) {
    const float* x   = (const float*)d_in[0];
    const int*   ei  = (const int*)  d_in[1];
    const float* W1  = (const float*)d_in[2];
    const float* as1 = (const float*)d_in[3];
    const float* ad1 = (const float*)d_in[4];
    const float* b1  = (const float*)d_in[5];
    const float* W2  = (const float*)d_in[6];
    const float* as2 = (const float*)d_in[7];
    const float* ad2 = (const float*)d_in[8];
    const float* b2  = (const float*)d_in[9];
    float* out = (float*)d_out;

    const int N = N_NODES;
    float* ws = (float*)d_ws;
    // workspace layout (float units)
    unsigned* m1   = (unsigned*)(ws);            // N*4
    float*    den1 = ws + 4 * N;                 // N*4
    float*    num1 = ws + 8 * N;                 // N*4
    float*    h2   = ws + 12 * N;                // N
    unsigned* m2   = (unsigned*)(ws + 13 * N);   // N
    float*    den2 = ws + 14 * N;                // N
    float*    num2 = ws + 15 * N;                // N
    float*    cst  = ws + 16 * N;                // 16

    // zero all accumulators (0 < fmap(-inf), a lower bound for the max maps)
    hipMemsetAsync(d_ws, 0, (size_t)(16 * N + 16) * sizeof(float), stream);

    k_precompute<<<1, 256, 0, stream>>>(W1, as1, ad1, cst);

    const int EB = (E2 + 255) / 256;   // 2048
    k_edge_max1<<<EB, 256, 0, stream>>>(ei, x, cst, m1);
    k_edge_sum1<<<EB, 256, 0, stream>>>(ei, x, cst, m1, den1, num1);

    // 128 nodes per block (8 waves x 16-node WMMA tiles); N = 1024 * 128 exactly
    k_h2_wmma<<<N / 128, 256, 0, stream>>>(num1, den1, W1, b1, W2, h2);

    k_edge_max2<<<EB, 256, 0, stream>>>(ei, h2, as2, ad2, m2);
    k_edge_sum2<<<EB, 256, 0, stream>>>(ei, h2, as2, ad2, m2, den2, num2);

    k_final<<<(N + 255) / 256, 256, 0, stream>>>(num2, den2, b2, out);
}